// LengthRegulator_16441134809203
// MI455X (gfx1250) — compile-verified
//
#include <hip/hip_runtime.h>

#define B_  16
#define TX_ 512
#define TY_ 4096
#define C_  192

typedef float v2f __attribute__((ext_vector_type(2)));
typedef float v4f __attribute__((ext_vector_type(4)));
typedef float v8f __attribute__((ext_vector_type(8)));

// ---------------------------------------------------------------------------
// Kernel 1: per-batch inclusive scan of durations -> int cum[B][TX]
// ---------------------------------------------------------------------------
__global__ __launch_bounds__(TX_) void cumsum_kernel(const float* __restrict__ dur,
                                                     int* __restrict__ cum) {
  __shared__ float s[TX_];
  const int b = blockIdx.x;
  const int t = threadIdx.x;
  s[t] = dur[b * TX_ + t];
  __syncthreads();
  #pragma unroll
  for (int off = 1; off < TX_; off <<= 1) {
    float v   = s[t];
    float add = (t >= off) ? s[t - off] : 0.0f;
    __syncthreads();
    s[t] = v + add;
    __syncthreads();
  }
  cum[b * TX_ + t] = (int)(s[t] + 0.5f);   // exact integer durations
}

// ---------------------------------------------------------------------------
// Kernel 2: attn[b,t,j] = (cum[t-1] <= j < cum[t]) * x_mask[b,t] * y_mask[b,j]
// 1024 j per 256-thread block; blocks whose j-range misses [lo,hi) store
// zeros without reading y_mask. Non-temporal 16B stores (write-once data).
// ---------------------------------------------------------------------------
__global__ __launch_bounds__(256) void attn_kernel(const int* __restrict__ cum,
                                                   const float* __restrict__ x_mask,
                                                   const float* __restrict__ y_mask,
                                                   float* __restrict__ attn) {
  const int b  = blockIdx.z;
  const int t  = blockIdx.y;
  const int jb = blockIdx.x * 1024;                 // block-uniform j base
  const int j0 = jb + threadIdx.x * 4;

  const int hi = cum[b * TX_ + t];
  const int lo = (t > 0) ? cum[b * TX_ + t - 1] : 0;

  v4f* dst = (v4f*)(attn + (size_t)(b * TX_ + t) * TY_ + j0);

  if (hi <= jb || lo >= jb + 1024) {                // uniform: nothing lands here
    v4f z = {0.0f, 0.0f, 0.0f, 0.0f};
    __builtin_nontemporal_store(z, dst);
    return;
  }

  const float xm = x_mask[b * TX_ + t];
  const float* ym = y_mask + b * TY_ + j0;
  v4f v;
  v.x = (j0 + 0 >= lo && j0 + 0 < hi) ? xm * ym[0] : 0.0f;
  v.y = (j0 + 1 >= lo && j0 + 1 < hi) ? xm * ym[1] : 0.0f;
  v.z = (j0 + 2 >= lo && j0 + 2 < hi) ? xm * ym[2] : 0.0f;
  v.w = (j0 + 3 >= lo && j0 + 3 < hi) ? xm * ym[3] : 0.0f;
  __builtin_nontemporal_store(v, dst);
}

// ---------------------------------------------------------------------------
// Kernel 3: mu_y[b] = attn[b]^T @ x[b] with V_WMMA_F32_16X16X4_F32.
// 4 waves/block, each wave owns one 16-row j-tile. cum & x_mask rows staged
// in LDS; K(t)-band per tile (<=16 wide, LDS binary search) walked branch-
// free. All 24 B-fragment scalars are loaded into live locals first so the
// compiler issues one load clause + one wait, then 12 back-to-back wmmas.
// ---------------------------------------------------------------------------
__global__ __launch_bounds__(128) void muy_wmma_kernel(const int* __restrict__ cum,
                                                       const float* __restrict__ x,
                                                       const float* __restrict__ x_mask,
                                                       const float* __restrict__ y_mask,
                                                       float* __restrict__ mu_y) {
  __shared__ int   scum[TX_];
  __shared__ float sxm[TX_];

  const int b   = blockIdx.y;
  const int tid = threadIdx.x;

  #pragma unroll
  for (int i = tid; i < TX_; i += 128) {
    scum[i] = cum[b * TX_ + i];
    sxm[i]  = x_mask[b * TX_ + i];
  }
  __syncthreads();

  const int wave = tid >> 5;
  const int lane = tid & 31;
  const int h    = lane >> 4;                 // lane half
  const int n    = lane & 15;                 // N index / A-row index
  const int J0   = blockIdx.x * 64 + wave * 16;

  const float* xb = x + (size_t)b * TX_ * C_;
  const int    j  = J0 + n;                   // this lane's A-matrix row (M = n)
  const float  ym = y_mask[b * TY_ + j];

  v8f acc[12];
  #pragma unroll
  for (int i = 0; i < 12; ++i) acc[i] = v8f{0.f,0.f,0.f,0.f,0.f,0.f,0.f,0.f};

  // upper_bound in LDS: smallest t with scum[t] > v (or TX_)
  int lo0 = 0, hi0 = TX_, lo1 = 0, hi1 = TX_;
  #pragma unroll
  for (int it = 0; it < 9; ++it) {            // 2^9 = 512
    int m0 = (lo0 + hi0) >> 1;
    int m1 = (lo1 + hi1) >> 1;
    int c0 = scum[m0], c1 = scum[m1];
    if (c0 > J0)      hi0 = m0; else lo0 = m0 + 1;
    if (c1 > J0 + 15) hi1 = m1; else lo1 = m1 + 1;
  }
  const int tlo = lo0;
  const int thi = (lo1 < TX_) ? lo1 : TX_ - 1;

  if (tlo < TX_) {
    for (int tb = (tlo & ~3); tb <= thi; tb += 4) {
      // ---- B fragments (x tile, 4x16 f32): K = 2*v + laneHalf ----
      // Issue all 24 independent loads first (one clause, one wait).
      // Out-of-range K columns have A == 0, so clamped rows need no select.
      const int tb0 = tb + h;                 // v = 0 row
      const int tb1 = tb + 2 + h;             // v = 1 row
      const float* r0 = xb + (size_t)((tb0 < TX_) ? tb0 : TX_ - 1) * C_ + n;
      const float* r1 = xb + (size_t)((tb1 < TX_) ? tb1 : TX_ - 1) * C_ + n;

      float bx[12], by[12];
      #pragma unroll
      for (int ct = 0; ct < 12; ++ct) {
        bx[ct] = r0[ct * 16];
        by[ct] = r1[ct * 16];
      }

      // ---- A fragment (attn^T tile, 16x4 f32): K = 2*laneHalf + v ----
      // Built from LDS while the global loads are in flight.
      const int ta0  = tb + 2 * h;
      const int ta1  = ta0 + 1;
      const int ta0c = (ta0 < TX_) ? ta0 : TX_ - 1;
      const int ta1c = (ta1 < TX_) ? ta1 : TX_ - 1;

      const int chi0 = scum[ta0c];
      const int chi1 = scum[ta1c];
      int clo0 = scum[(ta0c > 0) ? ta0c - 1 : 0];
      clo0 = (ta0c > 0) ? clo0 : 0;
      const int clo1 = scum[ta1c > 0 ? ta1c - 1 : 0];  // ta1 >= 1 always
      const float xm0 = sxm[ta0c];
      const float xm1 = sxm[ta1c];

      v2f a;
      a.x = (ta0 < TX_ && j >= clo0 && j < chi0) ? xm0 * ym : 0.0f;
      a.y = (ta1 < TX_ && j >= clo1 && j < chi1) ? xm1 * ym : 0.0f;

      // ---- 12 independent accumulating WMMAs ----
      #pragma unroll
      for (int ct = 0; ct < 12; ++ct) {
        v2f bf;
        bf.x = bx[ct];
        bf.y = by[ct];
        acc[ct] = __builtin_amdgcn_wmma_f32_16x16x4_f32(
            /*neg_a=*/false, a, /*neg_b=*/false, bf,
            /*c_mod=*/(short)0, acc[ct], /*reuse_a=*/false, /*reuse_b=*/false);
      }
    }
  }

  // ---- store C/D: VGPR r holds M = r + 8*laneHalf, N = lane&15 ----
  #pragma unroll
  for (int r = 0; r < 8; ++r) {
    const int M = r + 8 * h;
    float* row = mu_y + ((size_t)b * TY_ + J0 + M) * C_ + n;
    #pragma unroll
    for (int ct = 0; ct < 12; ++ct)
      __builtin_nontemporal_store(acc[ct][r], row + ct * 16);
  }
}

// ---------------------------------------------------------------------------
extern "C" void kernel_launch(void* const* d_in, const int* in_sizes, int n_in,
                              void* d_out, int out_size, void* d_ws, size_t ws_size,
                              hipStream_t stream) {
  // setup_inputs order: x, x_lengths, x_mask, y, y_lengths, y_mask, logw, durations
  const float* x      = (const float*)d_in[0];
  const float* x_mask = (const float*)d_in[2];
  const float* y_mask = (const float*)d_in[5];
  const float* dur    = (const float*)d_in[7];

  float* mu_y = (float*)d_out;                                  // (B, TY, C)
  float* attn = mu_y + (size_t)B_ * TY_ * C_;                   // (B, TX, TY)
  int*   cum  = (int*)d_ws;                                     // B*TX ints

  cumsum_kernel<<<B_, TX_, 0, stream>>>(dur, cum);

  attn_kernel<<<dim3(TY_ / 1024, TX_, B_), 256, 0, stream>>>(cum, x_mask, y_mask, attn);

  muy_wmma_kernel<<<dim3(TY_ / 64, B_), 128, 0, stream>>>(cum, x, x_mask, y_mask, mu_y);
}